// Pointer2D_87342454932158
// MI455X (gfx1250) — compile-verified
//
#include <hip/hip_runtime.h>
#include <hip/hip_bf16.h>

// ---------------------------------------------------------------------------
// Problem constants (from the reference)
// ---------------------------------------------------------------------------
#define BATCH 32
#define LSEQ  512
#define DHID  1024          // per-pointer hidden; embeddings last dim = 2*DHID
#define TWOD  2048
#define PBAND 4068          // L*A - A*(A-1)/2 with A=8
#define PFULL 4040          // rows 0..504 contribute 8 entries each

typedef __attribute__((ext_vector_type(2))) float v2f;
typedef __attribute__((ext_vector_type(8))) float v8f;

// ---------------------------------------------------------------------------
// Kernel 1: fused dual matvec  s[r] = <emb[r, 0:1024],    W>
//                              e[r] = <emb[r, 1024:2048], W>
// via V_WMMA_F32_16X16X4_F32. One wave32 owns a 16-row tile; W is replicated
// across the 16 B-columns so every column of D holds the same dot product.
// Memory-bound: 128 MiB streamed once; WMMA compute waste is free.
// ---------------------------------------------------------------------------
__global__ __launch_bounds__(256) void proj_wmma_kernel(
    const float* __restrict__ emb,   // [BATCH*LSEQ, TWOD]
    const float* __restrict__ W,     // [DHID]
    float* __restrict__ s_arr,       // [BATCH*LSEQ]
    float* __restrict__ e_arr)       // [BATCH*LSEQ]
{
    const int lane = threadIdx.x & 31;
    const int wave = threadIdx.x >> 5;                 // 8 waves / block
    const int tile = blockIdx.x * 8 + wave;            // 0..1023, 16 rows each
    const int m    = lane & 15;                        // A-matrix row within tile
    const int hi   = lane >> 4;                        // 0: K=0,1   1: K=2,3

    const long rowBase = (long)tile * 16;
    // A operand per ISA layout: lane(lo) -> A[m, k0+0..1], lane(hi) -> A[m, k0+2..3]
    const float* aRow  = emb + (rowBase + m) * (long)TWOD + 2 * hi;
    // B operand: VGPR r, half hi -> K = r + 2*hi; replicate W[k] over all 16 cols
    const float* wBase = W + 2 * hi;

    v8f cs = {};
    v8f ce = {};

#pragma unroll 4
    for (int k0 = 0; k0 < DHID; k0 += 4) {
        v2f a_s = *(const v2f*)(aRow + k0);            // start half
        v2f a_e = *(const v2f*)(aRow + k0 + DHID);     // end half
        v2f wv  = *(const v2f*)(wBase + k0);           // W chunk (bcast over N)
        // D = A*B + C ; 8-arg form: (neg_a, A, neg_b, B, c_mod, C, reuse_a, reuse_b)
        cs = __builtin_amdgcn_wmma_f32_16x16x4_f32(false, a_s, false, wv,
                                                   (short)0, cs, false, false);
        ce = __builtin_amdgcn_wmma_f32_16x16x4_f32(false, a_e, false, wv,
                                                   (short)0, ce, false, false);
    }

    // C/D layout: VGPR r = row r (lanes 0-15) / row r+8 (lanes 16-31), all
    // columns identical (B replicated) -> lanes 0 and 16 emit the 16 results.
    if (m == 0) {
        const long base = rowBase + 8 * hi;
#pragma unroll
        for (int r = 0; r < 8; ++r) {
            s_arr[base + r] = cs[r];
            e_arr[base + r] = ce[r];
        }
    }
}

// ---------------------------------------------------------------------------
// Kernel 2: banded gather + mask + softmax. One block per batch row.
// Band (row-major like np.nonzero): p < 4040 -> i = p/8, j = i + p%8;
// tail rows 505..511 contribute 7..1 entries.
// ---------------------------------------------------------------------------
__global__ __launch_bounds__(512) void band_softmax_kernel(
    const float* __restrict__ s_arr,
    const float* __restrict__ e_arr,
    const unsigned char* __restrict__ mask,   // [BATCH, LSEQ] bool (1 byte)
    const float* __restrict__ bias,           // [1]
    float* __restrict__ out)                  // [BATCH, PBAND]
{
    __shared__ float lg_lds[PBAND];
    __shared__ float red[16];

    const int b   = blockIdx.x;
    const int tid = threadIdx.x;
    const float bv = bias[0];
    const float* sb = s_arr + b * LSEQ;
    const float* eb = e_arr + b * LSEQ;
    const unsigned char* mb = mask + b * LSEQ;

    // Pass 1: logits + local max
    float lmax = -3.0e38f;
    for (int p = tid; p < PBAND; p += 512) {
        int i, j;
        if (p < PFULL) {
            i = p >> 3;
            j = i + (p & 7);
        } else {
            int r = p - PFULL;       // tail: rows 505..511, counts 7..1
            int i0 = 505, c = 7;
            while (r >= c) { r -= c; --c; ++i0; }
            i = i0;
            j = i0 + r;
        }
        float lg = sb[i] + eb[j] + bv;
        if (!(mb[i] && mb[j])) lg -= 1.0e7f;
        lg_lds[p] = lg;
        lmax = fmaxf(lmax, lg);
    }

    // Block max reduction (wave32)
    for (int off = 16; off; off >>= 1)
        lmax = fmaxf(lmax, __shfl_xor(lmax, off, 32));
    if ((tid & 31) == 0) red[tid >> 5] = lmax;
    __syncthreads();
    if (tid < 16) {
        float v = red[tid];
        for (int off = 8; off; off >>= 1)
            v = fmaxf(v, __shfl_xor(v, off, 32));
        if (tid == 0) red[0] = v;
    }
    __syncthreads();
    const float gmax = red[0];
    __syncthreads();

    // Pass 2: exponentiate (store back to LDS) + local sum
    float lsum = 0.0f;
    for (int p = tid; p < PBAND; p += 512) {
        float ev = expf(lg_lds[p] - gmax);
        lg_lds[p] = ev;
        lsum += ev;
    }

    // Block sum reduction
    for (int off = 16; off; off >>= 1)
        lsum += __shfl_xor(lsum, off, 32);
    if ((tid & 31) == 0) red[tid >> 5] = lsum;
    __syncthreads();
    if (tid < 16) {
        float v = red[tid];
        for (int off = 8; off; off >>= 1)
            v += __shfl_xor(v, off, 32);
        if (tid == 0) red[0] = v;
    }
    __syncthreads();
    const float inv = 1.0f / red[0];

    // Pass 3: normalize + write
    for (int p = tid; p < PBAND; p += 512)
        out[(long)b * PBAND + p] = lg_lds[p] * inv;
}

// ---------------------------------------------------------------------------
// Launch: inputs in setup_inputs() order: embeddings, mask, W, b
// ---------------------------------------------------------------------------
extern "C" void kernel_launch(void* const* d_in, const int* in_sizes, int n_in,
                              void* d_out, int out_size, void* d_ws, size_t ws_size,
                              hipStream_t stream) {
    const float*         emb  = (const float*)d_in[0];          // [32,512,2048] f32
    const unsigned char* mask = (const unsigned char*)d_in[1];  // [32,512] bool
    const float*         W    = (const float*)d_in[2];          // [1024,1] f32
    const float*         bias = (const float*)d_in[3];          // [1] f32
    float*               out  = (float*)d_out;                  // [32,4068] f32

    float* s_arr = (float*)d_ws;                 // [BATCH*LSEQ]
    float* e_arr = s_arr + BATCH * LSEQ;         // [BATCH*LSEQ]  (128 KB total)

    // 1024 row-tiles of 16, 8 waves (256 threads) per block -> 128 blocks.
    proj_wmma_kernel<<<128, 256, 0, stream>>>(emb, W, s_arr, e_arr);
    band_softmax_kernel<<<BATCH, 512, 0, stream>>>(s_arr, e_arr, mask, bias, out);
}